// CLS_1889785610440
// MI455X (gfx1250) — compile-verified
//
#include <hip/hip_runtime.h>
#include <math.h>

#define Bv  128          // batch
#define Ev  256          // embed channels
#define Mv  64           // positions
#define NTOT (Bv*Mv*Ev)  // 2,097,152 sequences per scale
#define BE  (Bv*Ev)      // 32768
#define CLS 5
#define EPSV 1e-5f

typedef float v2f __attribute__((ext_vector_type(2)));
typedef float v8f __attribute__((ext_vector_type(8)));

__device__ __forceinline__ float rnn_step(float x, float h, float w, float u, float bb) {
    // tanh(x*wih + bih + h*whh + bhh), with bb = bih+bhh folded
    return tanhf(fmaf(x, w, fmaf(h, u, bb)));
}

// ---------------------------------------------------------------------------
// Kernel 1: per-scale scalar RNNs (T=4,8,16) + conv(1x3) fusion.
// One thread per sequence index n = (b*M + m)*E + e.
// Writes fusion in [m][b*E+e] layout so kernel 2 scans coalesced.
// Dominant cost of the whole net: 235MB streamed -> ~10us at 23.3 TB/s.
// ---------------------------------------------------------------------------
__global__ void k1_rnn1_conv(const float* __restrict__ a0,
                             const float* __restrict__ a1,
                             const float* __restrict__ a2,
                             const float* __restrict__ wih, const float* __restrict__ whh,
                             const float* __restrict__ bih, const float* __restrict__ bhh,
                             const float* __restrict__ cw,  const float* __restrict__ cb,
                             float* __restrict__ fusion) {
    int n = blockIdx.x * blockDim.x + threadIdx.x;
    if (n >= NTOT) return;

    const float w0 = wih[0], u0 = whh[0], b0 = bih[0] + bhh[0];
    const float w1 = wih[1], u1 = whh[1], b1 = bih[1] + bhh[1];
    const float w2 = wih[2], u2 = whh[2], b2 = bih[2] + bhh[2];

    // scale 0: T=4 (one float4 per thread, lanes fully coalesced)
    float4 x = ((const float4*)a0)[n];
    float h0 = 0.f;
    h0 = rnn_step(x.x, h0, w0, u0, b0);
    h0 = rnn_step(x.y, h0, w0, u0, b0);
    h0 = rnn_step(x.z, h0, w0, u0, b0);
    h0 = rnn_step(x.w, h0, w0, u0, b0);

    // scale 1: T=8 (2x float4)
    float h1 = 0.f;
    const float4* a1v = (const float4*)a1;
    #pragma unroll
    for (int q = 0; q < 2; ++q) {
        float4 y = a1v[n * 2 + q];
        h1 = rnn_step(y.x, h1, w1, u1, b1);
        h1 = rnn_step(y.y, h1, w1, u1, b1);
        h1 = rnn_step(y.z, h1, w1, u1, b1);
        h1 = rnn_step(y.w, h1, w1, u1, b1);
    }

    // scale 2: T=16 (4x float4)
    float h2 = 0.f;
    const float4* a2v = (const float4*)a2;
    #pragma unroll
    for (int q = 0; q < 4; ++q) {
        float4 z = a2v[n * 4 + q];
        h2 = rnn_step(z.x, h2, w2, u2, b2);
        h2 = rnn_step(z.y, h2, w2, u2, b2);
        h2 = rnn_step(z.z, h2, w2, u2, b2);
        h2 = rnn_step(z.w, h2, w2, u2, b2);
    }

    // conv over S=3
    float f = fmaf(h0, cw[0], fmaf(h1, cw[1], fmaf(h2, cw[2], cb[0])));

    int e = n & (Ev - 1);
    int m = (n >> 8) & (Mv - 1);
    int b = n >> 14;
    fusion[m * BE + b * Ev + e] = f;   // [m][b*E+e]
}

// ---------------------------------------------------------------------------
// Kernel 2: second scalar RNN over M=64 steps; one thread per (b,e) lane.
// fusion is [m][lane] so each step is a fully coalesced 128KB slab (L2-hot:
// the whole 8MB intermediate fits in the 192MB L2).
// ---------------------------------------------------------------------------
__global__ void k2_rnn2(const float* __restrict__ fusion,
                        const float* __restrict__ wihp, const float* __restrict__ whhp,
                        const float* __restrict__ bihp, const float* __restrict__ bhhp,
                        float* __restrict__ feat) {
    int t = blockIdx.x * blockDim.x + threadIdx.x;   // t = b*E + e
    const float w = wihp[0], u = whhp[0], bb = bihp[0] + bhhp[0];
    float h = 0.f;
    #pragma unroll 4
    for (int m = 0; m < Mv; ++m)
        h = rnn_step(fusion[m * BE + t], h, w, u, bb);
    feat[t] = h;   // feat[b][e]
}

// ---------------------------------------------------------------------------
// Kernel 3: BatchNorm1d (batch stats, biased var) + ReLU -> A matrix for GEMM.
// One block of 256 threads; thread e owns channel e (coalesced across e).
// ---------------------------------------------------------------------------
__global__ void k3_bn_relu(const float* __restrict__ feat,
                           const float* __restrict__ gamma, const float* __restrict__ beta,
                           float* __restrict__ reluxn) {
    int e = threadIdx.x;
    float s = 0.f, s2 = 0.f;
    for (int b = 0; b < Bv; ++b) {
        float v = feat[b * Ev + e];
        s += v; s2 += v * v;
    }
    float mean  = s  * (1.f / Bv);
    float var   = s2 * (1.f / Bv) - mean * mean;
    float scale = gamma[e] * rsqrtf(var + EPSV);
    float shift = beta[e] - mean * scale;
    for (int b = 0; b < Bv; ++b) {
        float v = fmaf(feat[b * Ev + e], scale, shift);
        reluxn[b * Ev + e] = v > 0.f ? v : 0.f;
    }
}

// ---------------------------------------------------------------------------
// Kernel 4: logits = A(128x256) @ W^T(256x16) via V_WMMA_F32_16X16X4_F32
// (fp32 matrix pipe, no precision loss), then bias + softmax over 5 classes.
// One block = 8 waves; wave w owns rows [16w, 16w+16).
//
// f32 16x4 A layout (ISA 7.12.2): lane L -> row M = L&15;
//   VGPR0 holds K = kbase + 2*(L>>4), VGPR1 holds K = kbase + 2*(L>>4) + 1
// -> each lane loads one float2 of consecutive K. B (4x16) mirrored: lane's
// column N = L&15, same K split across lane halves.
// D layout: acc[j] = D[rowbase + 8*(L>>4) + j][L&15].
//
// B columns 5..15 don't need zero padding: WMMA column n of D depends only on
// column n of B, and softmax reads only D columns 0..4. So lanes with l>=5
// load column 0 (valid address, result ignored) -> no EXEC divergence in the
// hot loop, loop body is 2x global_load_b64 + v_wmma.
// ---------------------------------------------------------------------------
__global__ void k4_fnn_wmma_softmax(const float* __restrict__ A,     // relu_xn 128x256
                                    const float* __restrict__ W,     // fnn_w 5x256 row-major
                                    const float* __restrict__ bias,  // 5
                                    float* __restrict__ out) {       // 128x5
    __shared__ float lds[Bv * 16];
    const int tid  = threadIdx.x;
    const int wv   = tid >> 5;        // wave id 0..7 -> row tile
    const int lane = tid & 31;
    const int half = lane >> 4;
    const int l    = lane & 15;
    const int rowbase = wv * 16;
    const int bcol = (l < CLS) ? l : 0;   // clamped column, branch-free

    const float* __restrict__ arow_p = A + (rowbase + l) * Ev + 2 * half;
    const float* __restrict__ bcol_p = W + bcol * Ev + 2 * half;

    v8f acc = {};
    #pragma unroll 8
    for (int kk = 0; kk < Ev / 4; ++kk) {
        v2f av = *(const v2f*)(arow_p + kk * 4);   // even index -> 8B aligned
        v2f bv = *(const v2f*)(bcol_p + kk * 4);
        acc = __builtin_amdgcn_wmma_f32_16x16x4_f32(
            /*neg_a=*/false, av, /*neg_b=*/false, bv,
            /*c_mod=*/(short)0, acc, /*reuse_a=*/false, /*reuse_b=*/false);
    }

    #pragma unroll
    for (int j = 0; j < 8; ++j)
        lds[(rowbase + half * 8 + j) * 16 + l] = acc[j];
    __syncthreads();

    if (tid < Bv) {
        float lg[CLS];
        float mx = -3.4e38f;
        #pragma unroll
        for (int c = 0; c < CLS; ++c) {
            lg[c] = lds[tid * 16 + c] + bias[c];
            mx = fmaxf(mx, lg[c]);
        }
        float sum = 0.f;
        #pragma unroll
        for (int c = 0; c < CLS; ++c) { lg[c] = expf(lg[c] - mx); sum += lg[c]; }
        float inv = 1.f / sum;
        #pragma unroll
        for (int c = 0; c < CLS; ++c) out[tid * CLS + c] = lg[c] * inv;
    }
}

// ---------------------------------------------------------------------------
extern "C" void kernel_launch(void* const* d_in, const int* in_sizes, int n_in,
                              void* d_out, int out_size, void* d_ws, size_t ws_size,
                              hipStream_t stream) {
    (void)in_sizes; (void)n_in; (void)out_size; (void)ws_size;

    const float* a0       = (const float*)d_in[0];
    const float* a1       = (const float*)d_in[1];
    const float* a2       = (const float*)d_in[2];
    const float* r1_wih   = (const float*)d_in[3];
    const float* r1_whh   = (const float*)d_in[4];
    const float* r1_bih   = (const float*)d_in[5];
    const float* r1_bhh   = (const float*)d_in[6];
    const float* conv_w   = (const float*)d_in[7];
    const float* conv_b   = (const float*)d_in[8];
    const float* r2_wih   = (const float*)d_in[9];
    const float* r2_whh   = (const float*)d_in[10];
    const float* r2_bih   = (const float*)d_in[11];
    const float* r2_bhh   = (const float*)d_in[12];
    const float* gamma    = (const float*)d_in[13];
    const float* beta     = (const float*)d_in[14];
    const float* fnn_w    = (const float*)d_in[15];
    const float* fnn_b    = (const float*)d_in[16];
    float*       out      = (float*)d_out;

    // workspace carve-up: fusion 8MB | feat 128KB | relu_xn 128KB
    float* fusion = (float*)d_ws;
    float* feat   = fusion + (size_t)BE * Mv;
    float* reluxn = feat + BE;

    k1_rnn1_conv<<<dim3(NTOT / 256), dim3(256), 0, stream>>>(
        a0, a1, a2, r1_wih, r1_whh, r1_bih, r1_bhh, conv_w, conv_b, fusion);

    k2_rnn2<<<dim3(BE / 256), dim3(256), 0, stream>>>(
        fusion, r2_wih, r2_whh, r2_bih, r2_bhh, feat);

    k3_bn_relu<<<dim3(1), dim3(256), 0, stream>>>(feat, gamma, beta, reluxn);

    k4_fnn_wmma_softmax<<<dim3(1), dim3(256), 0, stream>>>(reluxn, fnn_w, fnn_b, out);
}